// SPDManifoldPooling_41059887350427
// MI455X (gfx1250) — compile-verified
//
#include <hip/hip_runtime.h>

typedef __attribute__((ext_vector_type(2))) float v2f;
typedef __attribute__((ext_vector_type(8))) float v8f;

#define LD64 68   // padded row stride for 64x64 LDS tiles (bank spread)
#define LD32 36   // padded row stride for 32x32 LDS tiles

// ---------------------------------------------------------------------------
// 64x64 f32 GEMM, D = A*B with B SYMMETRIC, all operands in LDS. 16 waves,
// one 16x16 C-tile per wave, K accumulated in steps of 4 via
// V_WMMA_F32_16X16X4_F32.
// A-frag (16x4 f32): lanes 0-15 hold M=lane, K={0,1}; lanes 16-31 M=lane-16,
// K={2,3}  -> contiguous 2-float (b64) read per lane.
// B-frag (4x16): VGPR0 = rows K0/K2, VGPR1 = rows K1/K3, N = lane%16. Using
// B = B^T we read B[n][k] (row-contiguous) instead of B[k][n] -> b64 read.
// C/D (16x16 f32, 8 VGPRs): VGPR r -> M = r + 8*(lane>=16), N = lane%16.
// ---------------------------------------------------------------------------
__device__ __forceinline__ void mm64s(const float* __restrict__ A,
                                      const float* __restrict__ B,
                                      float* __restrict__ D,
                                      int wave, int lane) {
  const int tm = wave >> 2, tn = wave & 3;
  const int half = lane >> 4, nn = lane & 15;
  v8f acc = {};
  const float* arow = A + (tm * 16 + nn) * LD64;   // row M = tm*16+nn of A
  const float* brow = B + (tn * 16 + nn) * LD64;   // row N of B == col N (sym)
#pragma unroll
  for (int kt = 0; kt < 16; ++kt) {
    const int k0 = kt * 4 + half * 2;
    v2f a, b;
    a.x = arow[k0];
    a.y = arow[k0 + 1];
    b.x = brow[k0];
    b.y = brow[k0 + 1];
    acc = __builtin_amdgcn_wmma_f32_16x16x4_f32(false, a, false, b,
                                                (short)0, acc, false, false);
  }
#pragma unroll
  for (int r = 0; r < 8; ++r)
    D[(tm * 16 + half * 8 + r) * LD64 + tn * 16 + nn] = acc[r];
}

// 32x32 f32 GEMM (B symmetric), waves 0-3 each own one 16x16 tile.
__device__ __forceinline__ void mm32s(const float* __restrict__ A,
                                      const float* __restrict__ B,
                                      float* __restrict__ D,
                                      int wave, int lane) {
  if (wave >= 4) return;                 // whole-wave uniform: EXEC stays all-1s
  const int tm = wave >> 1, tn = wave & 1;
  const int half = lane >> 4, nn = lane & 15;
  v8f acc = {};
  const float* arow = A + (tm * 16 + nn) * LD32;
  const float* brow = B + (tn * 16 + nn) * LD32;
#pragma unroll
  for (int kt = 0; kt < 8; ++kt) {
    const int k0 = kt * 4 + half * 2;
    v2f a, b;
    a.x = arow[k0];
    a.y = arow[k0 + 1];
    b.x = brow[k0];
    b.y = brow[k0 + 1];
    acc = __builtin_amdgcn_wmma_f32_16x16x4_f32(false, a, false, b,
                                                (short)0, acc, false, false);
  }
#pragma unroll
  for (int r = 0; r < 8; ++r)
    D[(tm * 16 + half * 8 + r) * LD32 + tn * 16 + nn] = acc[r];
}

// ---------------------------------------------------------------------------
// One workgroup per batch element:
//   B = X/c (c = inf-norm)  -> 4 coupled Newton-Schulz sqrt levels -> B^(1/16)
//   log(X) = 16*Taylor8(log(I+E)) + log(c) I
//   2x2 avg pool -> 32x32, symmetrize
//   exp via scaling-squaring: 12-term Taylor + s squarings
// All RHS operands are polynomials in one symmetric matrix -> symmetric, so
// the transposed-read GEMMs above are valid throughout.
// ---------------------------------------------------------------------------
__global__ __launch_bounds__(512)
void spd_logpool_exp(const float* __restrict__ Xg, float* __restrict__ outg) {
  __shared__ float buf[4][64 * LD64];    // 4 x 64x64 working matrices
  __shared__ float sb[4][32 * LD32];     // 4 x 32x32 working matrices
  __shared__ float red[64];
  __shared__ float scal[2];              // [0]=log(c), [1]=2^-s
  __shared__ int   ssq[1];

  const int tid  = threadIdx.x;
  const int wave = tid >> 5;
  const int lane = tid & 31;
  const float4* Xin4 = (const float4*)(Xg + (size_t)blockIdx.x * 4096);
  float4*       Out4 = (float4*)(outg + (size_t)blockIdx.x * 1024);

  // ---- load X (64x64) into buf0, 128-bit global loads / LDS stores ----
  for (int q = tid; q < 1024; q += 512) {          // 1024 float4 = 4096 floats
    float4 v = Xin4[q];
    const int row = q >> 4, c4 = (q & 15) << 2;    // 16 float4 per row
    *(float4*)&buf[0][row * LD64 + c4] = v;
  }
  __syncthreads();

  // ---- inf-norm c; scale B = X / c ----
  if (tid < 64) {
    float s = 0.f;
    for (int j = 0; j < 64; ++j) s += fabsf(buf[0][tid * LD64 + j]);
    red[tid] = s;
  }
  __syncthreads();
  if (tid == 0) {
    float m = 1e-30f;
    for (int i = 0; i < 64; ++i) m = fmaxf(m, red[i]);
    red[0]  = 1.f / m;
    scal[0] = logf(m);
  }
  __syncthreads();
  const float invc = red[0];
  for (int e = tid; e < 4096; e += 512)
    buf[0][(e >> 6) * LD64 + (e & 63)] *= invc;
  __syncthreads();

  // ---- 4 levels of coupled Newton-Schulz matrix square root ----
  int bB = 0, o1 = 1, o2 = 2, o3 = 3;
  for (int lvl = 0; lvl < 4; ++lvl) {
    int y = bB, z = o1, t1 = o2, t2 = o3;
    for (int e = tid; e < 4096; e += 512) {
      int r = e >> 6, c = e & 63;
      buf[z][r * LD64 + c] = (r == c) ? 1.f : 0.f;     // Z = I
    }
    __syncthreads();
    const int niter = (lvl == 0) ? 10 : (lvl == 1) ? 7 : (lvl == 2) ? 6 : 5;
    for (int it = 0; it < niter; ++it) {
      mm64s(buf[z], buf[y], buf[t1], wave, lane);      // T = Z*Y   (Y sym)
      __syncthreads();
      for (int e = tid; e < 4096; e += 512) {          // W = (3I - T)/2
        int r = e >> 6, c = e & 63;
        float v = buf[t1][r * LD64 + c];
        buf[t1][r * LD64 + c] = 0.5f * (((r == c) ? 3.f : 0.f) - v);
      }
      __syncthreads();
      mm64s(buf[y], buf[t1], buf[t2], wave, lane);     // Ynew = Y*W (W sym)
      __syncthreads();
      mm64s(buf[t1], buf[z], buf[y], wave, lane);      // Znew = W*Z (Z sym)
      __syncthreads();
      int ny = t2, nz = y, nt2 = z;
      y = ny; z = nz; t2 = nt2;                        // rotate buffers
    }
    bB = y; o1 = z; o2 = t1; o3 = t2;
  }

  // ---- log(I+E) Taylor, E = B^(1/16) - I ----
  if (tid < 64) buf[bB][tid * LD64 + tid] -= 1.f;
  __syncthreads();
  const int Lb = o1;
  for (int e = tid; e < 4096; e += 512) {
    int i = (e >> 6) * LD64 + (e & 63);
    buf[Lb][i] = buf[bB][i];                           // L = E
  }
  __syncthreads();
  int pa = bB, ping = o2, pong = o3;
  for (int j = 2; j <= 8; ++j) {
    mm64s(buf[pa], buf[bB], buf[ping], wave, lane);    // E^j   (E sym)
    __syncthreads();
    const float cf = ((j & 1) ? 1.f : -1.f) / (float)j;
    for (int e = tid; e < 4096; e += 512) {
      int i = (e >> 6) * LD64 + (e & 63);
      buf[Lb][i] += cf * buf[ping][i];
    }
    __syncthreads();
    pa = ping; int t = ping; ping = pong; pong = t;
  }

  // ---- pool 2x2 (logX = 16*L + log(c) I folded in) ----
  const float logc = scal[0];
  for (int e = tid; e < 1024; e += 512) {
    int i = e >> 5, j = e & 31;
    const float* Lp = buf[Lb];
    float v = Lp[(2 * i) * LD64 + 2 * j]     + Lp[(2 * i) * LD64 + 2 * j + 1]
            + Lp[(2 * i + 1) * LD64 + 2 * j] + Lp[(2 * i + 1) * LD64 + 2 * j + 1];
    sb[0][i * LD32 + j] = 4.f * v + ((i == j) ? 0.5f * logc : 0.f);
  }
  __syncthreads();
  for (int e = tid; e < 1024; e += 512) {              // symmetrize -> sb1
    int i = e >> 5, j = e & 31;
    sb[1][i * LD32 + j] = 0.5f * (sb[0][i * LD32 + j] + sb[0][j * LD32 + i]);
  }
  __syncthreads();

  // ---- exp scaling: s with ||H||inf / 2^s <= 0.25 ----
  if (tid < 32) {
    float s = 0.f;
    for (int j = 0; j < 32; ++j) s += fabsf(sb[1][tid * LD32 + j]);
    red[tid] = s;
  }
  __syncthreads();
  if (tid == 0) {
    float m = 0.f;
    for (int i = 0; i < 32; ++i) m = fmaxf(m, red[i]);
    int s = 0;
    while (m > 0.25f && s < 14) { m *= 0.5f; ++s; }
    ssq[0]  = s;
    scal[1] = exp2f(-(float)s);
  }
  __syncthreads();
  const float sc = scal[1];
  for (int e = tid; e < 1024; e += 512) {              // A=sb0, R=I+A=sb2
    int i = e >> 5, j = e & 31;
    float a = sb[1][i * LD32 + j] * sc;
    sb[0][i * LD32 + j] = a;
    sb[2][i * LD32 + j] = a + ((i == j) ? 1.f : 0.f);
  }
  __syncthreads();

  // ---- exp Taylor terms j = 2..12 ----
  int pc = 0, pn = 1;
  float invf = 1.f;
  for (int j = 2; j <= 12; ++j) {
    mm32s(sb[pc], sb[0], sb[pn], wave, lane);          // A^j  (A sym)
    __syncthreads();
    invf /= (float)j;                                  // 1/j!
    for (int e = tid; e < 1024; e += 512) {
      int idx = (e >> 5) * LD32 + (e & 31);
      sb[2][idx] += invf * sb[pn][idx];
    }
    __syncthreads();
    pc = pn; pn = (pc == 1) ? 3 : 1;
  }

  // ---- s squarings ----
  const int s = ssq[0];
  int rb = 2, tb = 0;
  for (int q = 0; q < s; ++q) {
    mm32s(sb[rb], sb[rb], sb[tb], wave, lane);         // R sym
    __syncthreads();
    int t = rb; rb = tb; tb = t;
  }

  // ---- write 32x32 result, 128-bit stores ----
  for (int q = tid; q < 256; q += 512) {               // 256 float4 = 1024 floats
    const int row = q >> 3, c4 = (q & 7) << 2;         // 8 float4 per row
    Out4[q] = *(const float4*)&sb[rb][row * LD32 + c4];
  }
}

extern "C" void kernel_launch(void* const* d_in, const int* in_sizes, int n_in,
                              void* d_out, int out_size, void* d_ws, size_t ws_size,
                              hipStream_t stream) {
  (void)n_in; (void)out_size; (void)d_ws; (void)ws_size;
  const float* X = (const float*)d_in[0];
  float* out = (float*)d_out;
  const int batches = in_sizes[0] / 4096;   // 8192 matrices of 64x64
  spd_logpool_exp<<<dim3(batches), dim3(512), 0, stream>>>(X, out);
}